// GOTDEvaluation_52055003627821
// MI455X (gfx1250) — compile-verified
//
#include <hip/hip_runtime.h>
#include <hip/hip_bf16.h>
#include <stdint.h>

#define HM    64
#define ELEMS (HM * HM)     // 4096 floats = 16 KB per heatmap
#define MID   10

typedef unsigned int u32x4 __attribute__((ext_vector_type(4)));
typedef int          i32x8 __attribute__((ext_vector_type(8)));
typedef int          i32x4 __attribute__((ext_vector_type(4)));

struct Scratch {
    unsigned gval, hval;    // float-bits of max (clipped values are >= 0)
    unsigned gidx, hidx;    // argmax (first occurrence via atomicMin)
    unsigned conf;          // float-bits of region max
    unsigned col0, col1;    // 64-bit "any column >= 0.5" mask
    unsigned row0, row1;    // 64-bit "any row    >= 0.5" mask
};

__device__ __forceinline__ unsigned rfl(unsigned v) {
    return (unsigned)__builtin_amdgcn_readfirstlane((int)v);
}

// One TDM descriptor: 1-row tile of 4096 x 4-byte elements, global -> LDS.
__device__ __forceinline__ void tdm_tile_load(unsigned long long gaddr, unsigned ldsoff) {
    u32x4 g0;
    g0[0] = 1u;                                                   // count=1, user mode
    g0[1] = rfl(ldsoff);                                          // lds_addr (bytes)
    g0[2] = rfl((unsigned)(gaddr & 0xFFFFFFFFull));               // global_addr[31:0]
    g0[3] = rfl(((unsigned)((gaddr >> 32) & 0x01FFFFFFull))       // global_addr[56:32]
                | 0x80000000u);                                   // type=2 ("image")
    i32x8 g1;
    g1[0] = 0x00020000;             // wg_mask=0, data_size=2 (4 bytes), no flags
    g1[1] = (int)(4096u << 16);     // tensor_dim0[15:0] in bits [31:16]
    g1[2] = (int)(1u << 16);        // tensor_dim1 = 1
    g1[3] = (int)(4096u << 16);     // tile_dim0 = 4096
    g1[4] = 1;                      // tile_dim1 = 1, tile_dim2 = 0
    g1[5] = 4096;                   // tensor_dim0_stride[31:0]
    g1[6] = 0;
    g1[7] = 0;
    i32x4 z4 = {0, 0, 0, 0};        // groups 2/3 unused for 2D tensors
    i32x8 z8 = {0, 0, 0, 0, 0, 0, 0, 0};
    __builtin_amdgcn_tensor_load_to_lds(g0, g1, z4, z4, z8, 0);   // 6-arg form (clang-23)
}

__global__ void __launch_bounds__(256)
gotd_score_kernel(const float* __restrict__ gaze,
                  const float* __restrict__ head,
                  const float* __restrict__ connect,
                  const float* __restrict__ watch,
                  float* __restrict__ out) {
    extern __shared__ __align__(16) float dsm[];   // [2 slots][g|h][ELEMS]
    __shared__ Scratch scr[2];

    const int tid = threadIdx.x;
    const int e0  = blockIdx.x * 2;

    if (tid == 0) {
        #pragma unroll
        for (int s = 0; s < 2; ++s) {
            scr[s].gval = 0u; scr[s].hval = 0u;
            scr[s].gidx = 0xFFFFFFFFu; scr[s].hidx = 0xFFFFFFFFu;
            scr[s].conf = 0u;
            scr[s].col0 = scr[s].col1 = 0u;
            scr[s].row0 = scr[s].row1 = 0u;
        }
    }

    const unsigned dsm_off = (unsigned)(unsigned long long)(uintptr_t)(void*)dsm;

    // Wave 0 issues all four TDM DMAs (EXEC-independent, SGPR descriptors).
    if (tid < 32) {
        const unsigned long long gbase = (unsigned long long)(uintptr_t)gaze;
        const unsigned long long hbase = (unsigned long long)(uintptr_t)head;
        #pragma unroll
        for (int s = 0; s < 2; ++s) {
            const unsigned long long off = (unsigned long long)(e0 + s) * (ELEMS * 4ull);
            tdm_tile_load(gbase + off, dsm_off + (unsigned)((s * 2 + 0) * ELEMS * 4));
            tdm_tile_load(hbase + off, dsm_off + (unsigned)((s * 2 + 1) * ELEMS * 4));
        }
    }

    const int base = tid * 16;        // each thread owns 16 consecutive pixels
    const int y    = base >> 6;       // its row
    const int x0   = base & 63;       // its first column

    #pragma unroll
    for (int s = 0; s < 2; ++s) {
        if (tid < 32) {
            if (s == 0) __builtin_amdgcn_s_wait_tensorcnt(2);  // first 2 DMAs done (in-order)
            else        __builtin_amdgcn_s_wait_tensorcnt(0);  // all done
        }
        __syncthreads();

        const int e    = e0 + s;
        const int goff = (s * 2 + 0) * ELEMS;
        const int hoff = (s * 2 + 1) * ELEMS;

        float gv[16], hv[16];
        #pragma unroll
        for (int v = 0; v < 4; ++v) {
            float4 tg = *(const float4*)&dsm[goff + base + 4 * v];
            float4 th = *(const float4*)&dsm[hoff + base + 4 * v];
            gv[4*v+0] = tg.x; gv[4*v+1] = tg.y; gv[4*v+2] = tg.z; gv[4*v+3] = tg.w;
            hv[4*v+0] = th.x; hv[4*v+1] = th.y; hv[4*v+2] = th.z; hv[4*v+3] = th.w;
        }

        // ---- phase A: clip, local max, threshold masks ----
        float gmax = 0.f, hmax = 0.f;
        unsigned c0 = 0u, c1 = 0u; int rowany = 0;
        #pragma unroll
        for (int i = 0; i < 16; ++i) {
            float g = fminf(fmaxf(gv[i], 0.f), 1.f); gv[i] = g;
            float h = fminf(fmaxf(hv[i], 0.f), 1.f); hv[i] = h;
            gmax = fmaxf(gmax, g);
            hmax = fmaxf(hmax, h);
            if (h >= 0.5f) {
                const int x = x0 + i;
                if (x < 32) c0 |= (1u << x); else c1 |= (1u << (x - 32));
                rowany = 1;
            }
        }
        atomicMax(&scr[s].gval, __float_as_uint(gmax));
        atomicMax(&scr[s].hval, __float_as_uint(hmax));
        if (c0) atomicOr(&scr[s].col0, c0);
        if (c1) atomicOr(&scr[s].col1, c1);
        if (rowany) {
            if (y < 32) atomicOr(&scr[s].row0, 1u << y);
            else        atomicOr(&scr[s].row1, 1u << (y - 32));
        }
        __syncthreads();

        // ---- phase B: first-occurrence argmax indices ----
        const unsigned gvb = scr[s].gval;
        const unsigned hvb = scr[s].hval;
        #pragma unroll
        for (int i = 0; i < 16; ++i) {
            if (__float_as_uint(gv[i]) == gvb) atomicMin(&scr[s].gidx, (unsigned)(base + i));
            if (__float_as_uint(hv[i]) == hvb) atomicMin(&scr[s].hidx, (unsigned)(base + i));
        }
        __syncthreads();

        // ---- phase C: bbox + region max (head values still in VGPRs) ----
        const unsigned long long col =
            ((unsigned long long)scr[s].col1 << 32) | (unsigned long long)scr[s].col0;
        const unsigned long long row =
            ((unsigned long long)scr[s].row1 << 32) | (unsigned long long)scr[s].row0;
        const int hidx = (int)scr[s].hidx;
        const int py = hidx >> 6, px = hidx & 63;
        const bool has = (col != 0ull);
        const int x1 = has ? (__ffsll((unsigned long long)col) - 1) : px;
        const int x2 = has ? (64 - __clzll((long long)col))         : (px + 1);
        const int y1 = has ? (__ffsll((unsigned long long)row) - 1) : py;
        const int y2 = has ? (64 - __clzll((long long)row))         : (py + 1);

        float cmax = 0.f;
        if (y >= y1 && y < y2) {
            #pragma unroll
            for (int i = 0; i < 16; ++i) {
                const int x = x0 + i;
                if (x >= x1 && x < x2) cmax = fmaxf(cmax, hv[i]);
            }
        }
        atomicMax(&scr[s].conf, __float_as_uint(cmax));
        __syncthreads();

        // ---- final scalar tail: 10 sparse connect samples + score ----
        if (tid == 0) {
            const float conf_head = __uint_as_float(scr[s].conf);
            const float conf_gaze = __uint_as_float(gvb);
            const int   gidx = (int)scr[s].gidx;
            const float gpx = (float)(gidx & 63);
            const float gpy = (float)(gidx >> 6);
            const float cx  = 0.5f * (float)(x1 + x2);
            const float cy  = 0.5f * (float)(y1 + y2);
            const float hcx = fminf(fmaxf(rintf(cx), 0.f), 63.f);
            const float hcy = fminf(fmaxf(rintf(cy), 0.f), 63.f);
            const float* cp = connect + (size_t)e * ELEMS;
            float sum = 0.f;
            #pragma unroll
            for (int i = 0; i < MID; ++i) {
                const float t  = (float)i * (1.0f / 9.0f);
                const int   xi = (int)rintf(hcx + (gpx - hcx) * t);
                const int   yi = (int)rintf(hcy + (gpy - hcy) * t);
                const float cval = cp[yi * HM + xi];
                sum += fminf(fmaxf(cval, 0.f), 1.f);
            }
            const float vx = gpx - hcx, vy = gpy - hcy;
            const float nrm = sqrtf(vx * vx + vy * vy);
            const float dp  = fminf((0.5f * (float)HM) / nrm - 1.0f, 0.0f);
            const float swdp = sum * (1.0f / (float)MID) + dp;
            const float w = watch[e];
            out[e] = (w > 0.5f)
                   ? (conf_head + (1.0f - conf_gaze) - swdp) * (1.0f / 3.0f)
                   : (conf_head + conf_gaze + swdp) * (1.0f / 3.0f);
        }
        // next slot uses disjoint LDS buffers + scratch; its leading barrier suffices
    }
}

extern "C" void kernel_launch(void* const* d_in, const int* in_sizes, int n_in,
                              void* d_out, int out_size, void* d_ws, size_t ws_size,
                              hipStream_t stream) {
    const float* gaze    = (const float*)d_in[0];
    const float* head    = (const float*)d_in[1];
    const float* connect = (const float*)d_in[2];
    const float* watch   = (const float*)d_in[3];
    float* out = (float*)d_out;

    const int total  = in_sizes[3];        // B*Q = 8192 elements
    const int blocks = total / 2;          // 2 elements per workgroup (double-buffered)
    const size_t shmem = 2ull * 2ull * ELEMS * sizeof(float);  // 64 KB dynamic LDS

    gotd_score_kernel<<<blocks, 256, shmem, stream>>>(gaze, head, connect, watch, out);
}